// graph_augmentation_50474455662694
// MI455X (gfx1250) — compile-verified
//
#include <hip/hip_runtime.h>

// ---------------------------------------------------------------------------
// Problem constants (fixed by the reference setup)
// ---------------------------------------------------------------------------
#define BATCH   8192
#define NDIM    1024
#define GROUPS  3
#define ROWS    (GROUPS * BATCH)      // 24576 combined rows
#define TIME    6
#define GAMMA   0.9f
#define LDA     1028                  // padded LDS row stride (floats): 1028 % 64 == 4 -> conflict-free

typedef float v2f __attribute__((ext_vector_type(2)));
typedef float v8f __attribute__((ext_vector_type(8)));

// ---------------------------------------------------------------------------
// Kernel 1: R = sigmoid((bias + gumbel(U)) / 0.5) -> d_out[24576..]
//           also zero the vs region d_out[0..24575]
// ---------------------------------------------------------------------------
__global__ __launch_bounds__(256) void init_R_kernel(const float* __restrict__ fc1_b,
                                                     const float* __restrict__ U,
                                                     float* __restrict__ out)
{
    const size_t i = (size_t)blockIdx.x * blockDim.x + threadIdx.x;   // over BATCH*NDIM
    const int n = (int)(i & (NDIM - 1));
    const float u = U[i];
    const float gum = -logf(-logf(u + 1e-20f) + 1e-20f);
    const float z = (fc1_b[n] + gum) * 2.0f;                          // / TEMP(0.5)
    out[(size_t)ROWS + i] = 1.0f / (1.0f + expf(-z));                 // R
    if (i < (size_t)ROWS) out[i] = 0.0f;                              // zero vs
}

// ---------------------------------------------------------------------------
// Kernel 2 (per step): per-row  r = sum(s*R);  vs[b*3+g] += gamma * r;
//                      s <- s * (1 - R)   (in place; skipped on last step)
// One 256-thread block per combined row. step==0 reads x1/x2/x3 directly.
// ---------------------------------------------------------------------------
__global__ __launch_bounds__(256) void pre_step_kernel(const float* __restrict__ x1,
                                                       const float* __restrict__ x2,
                                                       const float* __restrict__ x3,
                                                       const float* __restrict__ R,
                                                       float* __restrict__ s,
                                                       float* __restrict__ vs,
                                                       float gamma, int step, int write_mask)
{
    const int rw = blockIdx.x;              // 0..ROWS-1
    const int g  = rw >> 13;                // rw / BATCH
    const int b  = rw & (BATCH - 1);

    const float* __restrict__ srow;
    if (step == 0)
        srow = (g == 0 ? x1 : (g == 1 ? x2 : x3)) + (size_t)b * NDIM;
    else
        srow = s + (size_t)rw * NDIM;

    const float* __restrict__ Rrow = R + (size_t)b * NDIM;
    float* __restrict__ orow = s + (size_t)rw * NDIM;

    const int c = threadIdx.x * 4;          // 256 threads * 4 = 1024 columns
    const float4 sv = *(const float4*)(srow + c);
    const float4 rv = *(const float4*)(Rrow + c);

    float acc = sv.x * rv.x + sv.y * rv.y + sv.z * rv.z + sv.w * rv.w;

    if (write_mask) {
        float4 o;
        o.x = sv.x * (1.0f - rv.x);
        o.y = sv.y * (1.0f - rv.y);
        o.z = sv.z * (1.0f - rv.z);
        o.w = sv.w * (1.0f - rv.w);
        *(float4*)(orow + c) = o;
    }

    // wave32 reduction, then cross-wave via LDS
    acc += __shfl_xor(acc, 1);
    acc += __shfl_xor(acc, 2);
    acc += __shfl_xor(acc, 4);
    acc += __shfl_xor(acc, 8);
    acc += __shfl_xor(acc, 16);

    __shared__ float wsum[8];
    const int lane = threadIdx.x & 31;
    const int wave = threadIdx.x >> 5;
    if (lane == 0) wsum[wave] = acc;
    __syncthreads();
    if (threadIdx.x == 0) {
        float r = 0.0f;
        #pragma unroll
        for (int w = 0; w < 8; ++w) r += wsum[w];
        vs[(size_t)b * GROUPS + g] += gamma * r;   // sequential launches: no atomics needed
    }
}

// ---------------------------------------------------------------------------
// Kernel 3 (per step, steps 0..4): logit = s @ W^T ; s <- |logit|/rowsum(logit)
// In place: one block per 16-row tile, full 1024 output columns per block.
// 8 waves; wave w owns column tiles (w*8+t)*16, t=0..7.  f32 WMMA 16x16x4.
// ---------------------------------------------------------------------------
__global__ __launch_bounds__(256) void step_gemm_kernel(float* __restrict__ s,
                                                        const float* __restrict__ W)
{
    __shared__ float Alds[16 * LDA];
    __shared__ float rpart[8][16];
    __shared__ float rtot[16];

    const int row0 = blockIdx.x * 16;
    const int tid  = threadIdx.x;

    // Stage A (16 rows x 1024 cols) into LDS, float4-coalesced.
    for (int i = tid; i < 16 * (NDIM / 4); i += 256) {
        const int r = i / (NDIM / 4);
        const int c = (i % (NDIM / 4)) * 4;
        const float4 v = *(const float4*)(s + (size_t)(row0 + r) * NDIM + c);
        float* dst = &Alds[r * LDA + c];
        dst[0] = v.x; dst[1] = v.y; dst[2] = v.z; dst[3] = v.w;
    }
    __syncthreads();

    const int lane = tid & 31;
    const int wave = tid >> 5;
    const int half = lane >> 4;      // 0 | 1
    const int l16  = lane & 15;

    v8f acc[8] = {};                 // 8 column tiles x 8 VGPRs

    // Per-tile W base: B fragment b[v] = W[col0+l16][k + v + 2*half]
    const float* wbase[8];
    #pragma unroll
    for (int t = 0; t < 8; ++t) {
        const int col = (wave * 8 + t) * 16 + l16;
        wbase[t] = W + (size_t)col * NDIM + 2 * half;
    }

    const float* abase = &Alds[l16 * LDA + 2 * half];

    #pragma unroll 2
    for (int k = 0; k < NDIM; k += 4) {
        const v2f a = *(const v2f*)(abase + k);            // ds_load_b64, conflict-free
        #pragma unroll
        for (int t = 0; t < 8; ++t) {
            const v2f b = *(const v2f*)(wbase[t] + k);     // global_load_b64 (L2-resident W)
            acc[t] = __builtin_amdgcn_wmma_f32_16x16x4_f32(
                false, a, false, b, (short)0, acc[t], false, false);
        }
    }

    // Signed row sums: D layout => VGPR i holds row (i + 8*half), col = l16 per tile.
    float psum[8];
    #pragma unroll
    for (int i = 0; i < 8; ++i) {
        float v = 0.0f;
        #pragma unroll
        for (int t = 0; t < 8; ++t) v += acc[t][i];
        v += __shfl_xor(v, 1);
        v += __shfl_xor(v, 2);
        v += __shfl_xor(v, 4);
        v += __shfl_xor(v, 8);       // sum over the 16 columns of this half
        psum[i] = v;
    }
    if (l16 == 0) {
        #pragma unroll
        for (int i = 0; i < 8; ++i) rpart[wave][i + 8 * half] = psum[i];
    }
    __syncthreads();
    if (tid < 16) {
        float v = 0.0f;
        #pragma unroll
        for (int w = 0; w < 8; ++w) v += rpart[w][tid];
        rtot[tid] = v;
    }
    __syncthreads();

    float rinv[8];
    #pragma unroll
    for (int i = 0; i < 8; ++i) rinv[i] = 1.0f / rtot[i + 8 * half];

    // Normalize and write back in place (row blocks are independent).
    #pragma unroll
    for (int t = 0; t < 8; ++t) {
        const int col = (wave * 8 + t) * 16 + l16;
        #pragma unroll
        for (int i = 0; i < 8; ++i) {
            const int row = row0 + i + 8 * half;
            s[(size_t)row * NDIM + col] = fabsf(acc[t][i]) * rinv[i];
        }
    }
}

// ---------------------------------------------------------------------------
// Launch: R/vs-init, then 6x pre_step (vg update + mask) interleaved with
// 5x WMMA gemm+normalize (last step's gemm output is dead -> skipped).
// ---------------------------------------------------------------------------
extern "C" void kernel_launch(void* const* d_in, const int* in_sizes, int n_in,
                              void* d_out, int out_size, void* d_ws, size_t ws_size,
                              hipStream_t stream) {
    const float* x1    = (const float*)d_in[0];
    const float* x2    = (const float*)d_in[1];
    const float* x3    = (const float*)d_in[2];
    const float* W     = (const float*)d_in[3];
    const float* fc1_b = (const float*)d_in[4];
    const float* U     = (const float*)d_in[5];
    (void)in_sizes; (void)n_in; (void)out_size; (void)ws_size;

    float* out = (float*)d_out;
    float* vs  = out;                 // [BATCH, 3]
    float* R   = out + ROWS;          // [BATCH, NDIM]
    float* s   = (float*)d_ws;        // [ROWS, NDIM] workspace (96 MB)

    init_R_kernel<<<(BATCH * NDIM) / 256, 256, 0, stream>>>(fc1_b, U, out);

    float gamma = 1.0f;
    for (int t = 0; t < TIME; ++t) {
        pre_step_kernel<<<ROWS, 256, 0, stream>>>(x1, x2, x3, R, s, vs, gamma, t,
                                                  (t < TIME - 1) ? 1 : 0);
        if (t < TIME - 1)
            step_gemm_kernel<<<ROWS / 16, 256, 0, stream>>>(s, W);
        gamma *= GAMMA;
    }
}